// ChainRelativePositionEmbedding_44375602102601
// MI455X (gfx1250) — compile-verified
//
#include <hip/hip_runtime.h>
#include <hip/hip_bf16.h>

typedef __attribute__((ext_vector_type(2))) float v2f;
typedef __attribute__((ext_vector_type(8))) float v8f;
typedef __attribute__((ext_vector_type(4))) float f4_t;

// ---------------------------------------------------------------------------
// Kernel 1: reconstruct per-chain residue indices from asym_id.
// idx[i] = i - (start of i's chain). Chains are contiguous runs of asym_id.
// ---------------------------------------------------------------------------
__global__ void crpe_compute_idx(const int* __restrict__ asym,
                                 int* __restrict__ idx, int L) {
  int i = blockIdx.x * blockDim.x + threadIdx.x;
  if (i >= L) return;
  int a = asym[i];
  int s = i;
  while (s > 0 && asym[s - 1] == a) s--;   // short walk, all cache hits
  idx[i] = i - s;
}

// ---------------------------------------------------------------------------
// Kernel 2: build the 66-row table with WMMA:  table = E @ Wt + bias
//   E is [R x K] selection matrix (R = 2*RM+2, K = 2*RM+3):
//     row r < R-1 (same chain, ridx_clip=r): e[0]=1, e[1+r]=1
//     row R-1     (cross chain):             e[K-1]=1
//   Wt[k][n] = weight[n*K + k]  (weight is [nd, K] row-major)
// One wave per 16x16 output tile; V_WMMA_F32_16X16X4_F32, k-steps of 4.
// Exact in f32 (each dot product has <=2 nonzero terms).
// ---------------------------------------------------------------------------
__device__ __forceinline__ float crpe_E(int r, int c, int R, int K) {
  if (c >= K || r >= R) return 0.0f;
  if (r < R - 1) return (c == r + 1 || c == 0) ? 1.0f : 0.0f;
  return (c == K - 1) ? 1.0f : 0.0f;
}

__device__ __forceinline__ float crpe_Wt(const float* __restrict__ W,
                                         int k, int n, int K) {
  int kk = (k < K) ? k : 0;          // clamped, unconditionally safe load
  float v = W[(size_t)n * K + kk];
  return (k < K) ? v : 0.0f;
}

__global__ __launch_bounds__(32)
void crpe_table_wmma(const float* __restrict__ W, const float* __restrict__ bias,
                     float* __restrict__ table, int R, int K, int nd) {
  const int lane = threadIdx.x & 31;
  const int hi   = lane >> 4;        // 0: lanes 0-15, 1: lanes 16-31
  const int lo   = lane & 15;
  const int mT = blockIdx.y, nT = blockIdx.x;
  const int m = mT * 16 + lo;        // A row (lanes 0-15 and 16-31 both M=0..15)
  const int n = nT * 16 + lo;        // B/C/D column
  const int Kpad = (K + 3) & ~3;

  v8f acc = {};
  for (int k0 = 0; k0 < Kpad; k0 += 4) {
    const int ka = k0 + 2 * hi;      // VGPR0 holds K={0,2}, VGPR1 holds K={1,3}
    v2f a, b;
    a[0] = crpe_E(m, ka,     R, K);
    a[1] = crpe_E(m, ka + 1, R, K);
    b[0] = crpe_Wt(W, ka,     n, K);
    b[1] = crpe_Wt(W, ka + 1, n, K);
    acc = __builtin_amdgcn_wmma_f32_16x16x4_f32(
        /*neg_a=*/false, a, /*neg_b=*/false, b,
        /*c_mod=*/(short)0, acc, /*reuse_a=*/false, /*reuse_b=*/false);
  }

  const float bs = bias[n];
#pragma unroll
  for (int v = 0; v < 8; v++) {      // C/D: VGPR v -> row M = v (+8 upper half)
    int row = mT * 16 + v + 8 * hi;
    if (row < R) table[(size_t)row * nd + n] = acc[v] + bs;
  }
}

// ---------------------------------------------------------------------------
// Kernel 3: bandwidth-limited gather-broadcast.
// Table (R*nd floats, 33.8KB) lives in LDS. Each wave handles 32 consecutive
// j for a fixed i; per iteration the wave-uniform table row is picked and all
// 32 lanes copy one float4 each: ds_load_b128 -> NT global_store_b128.
// ---------------------------------------------------------------------------
__global__ __launch_bounds__(256)
void crpe_emit(const int* __restrict__ asym, const int* __restrict__ idx,
               const float* __restrict__ table, float* __restrict__ out,
               int L, int RM, int nd) {
  extern __shared__ float smem[];
  const int nd4 = nd >> 2;
  const int R = 2 * RM + 2;

  // cooperative table load into LDS (float4 granularity)
  {
    const f4_t* t4 = (const f4_t*)table;
    f4_t* s4 = (f4_t*)smem;
    const int total4 = R * nd4;
    for (int t = threadIdx.x; t < total4; t += blockDim.x) s4[t] = t4[t];
  }
  __syncthreads();

  const int lane = threadIdx.x & 31;
  const int wid  = threadIdx.x >> 5;
  const int i = blockIdx.y;
  const int jbase = blockIdx.x * 256 + wid * 32;
  if (jbase >= L) return;

  const int a_i = asym[i];
  const int x_i = idx[i];

  // each lane preloads the j-metadata for its own j; broadcast per iteration
  const int jl = jbase + lane;
  int a_jl = 0, x_jl = 0;
  if (jl < L) { a_jl = asym[jl]; x_jl = idx[jl]; }

  const f4_t* s4 = (const f4_t*)smem;
  f4_t* out4 = (f4_t*)out;
  const size_t rowbase = ((size_t)i * L + jbase) * nd4;
  const int jcnt = (L - jbase < 32) ? (L - jbase) : 32;

  for (int s = 0; s < jcnt; s++) {
    const int a_j = __shfl(a_jl, s);
    const int x_j = __shfl(x_jl, s);
    int rr = x_i - x_j + RM;
    rr = rr < 0 ? 0 : (rr > 2 * RM ? 2 * RM : rr);
    const int r = (a_j == a_i) ? rr : (2 * RM + 1);

    const f4_t* src = s4 + (size_t)r * nd4;
    f4_t* dst = out4 + rowbase + (size_t)s * nd4;
    for (int q = lane; q < nd4; q += 32)          // nd=128 -> exactly one f4/lane
      __builtin_nontemporal_store(src[q], dst + q);
  }
}

// ---------------------------------------------------------------------------
extern "C" void kernel_launch(void* const* d_in, const int* in_sizes, int n_in,
                              void* d_out, int out_size, void* d_ws, size_t ws_size,
                              hipStream_t stream) {
  // inputs: [0] lengths(int64,[3])  [1] asym_id(int32,[L])
  //         [2] weight(f32,[nd,K])  [3] bias(f32,[nd])
  const int L  = in_sizes[1];
  const int nd = in_sizes[3];
  const int K  = in_sizes[2] / nd;        // 2*RM + 3 = 67
  const int RM = (K - 3) / 2;             // 32
  const int R  = 2 * RM + 2;              // 66 unique table rows

  const int*   asym = (const int*)d_in[1];
  const float* W    = (const float*)d_in[2];
  const float* bias = (const float*)d_in[3];
  float*       out  = (float*)d_out;

  // workspace layout: [table: R*nd f32][align 256][idx: L i32]
  float* table = (float*)d_ws;
  size_t table_bytes = (size_t)R * nd * sizeof(float);
  int* idx = (int*)((char*)d_ws + ((table_bytes + 255) & ~(size_t)255));

  // 1) residue indices
  crpe_compute_idx<<<(L + 255) / 256, 256, 0, stream>>>(asym, idx, L);

  // 2) 66-row table via f32 WMMA (grid: 8 N-tiles x 5 M-tiles, 1 wave each)
  dim3 tgrid(nd / 16, (R + 15) / 16);
  crpe_table_wmma<<<tgrid, 32, 0, stream>>>(W, bias, table, R, K, nd);

  // 3) stream 1.21 GB of output rows from the LDS-resident table
  dim3 egrid((L + 255) / 256, L);
  size_t smem = (size_t)R * nd * sizeof(float);   // 33,792 B
  crpe_emit<<<egrid, 256, smem, stream>>>(asym, idx, table, out, L, RM, nd);
}